// FinalZPrjs_Conv_Layer_35940286333046
// MI455X (gfx1250) — compile-verified
//
#include <hip/hip_runtime.h>
#include <hip/hip_bf16.h>

typedef __attribute__((ext_vector_type(2))) float v2f;
typedef __attribute__((ext_vector_type(8))) float v8f;

#define BATCH 8
#define CH    16
#define DHW   (96*96*96)          // 884736 spatial per (b, channel)
#define PAIRS (DHW/32)            // 27648 32-wide spatial tile-pairs per batch

// ---------------------------------------------------------------------------
// Kernel 1: fold z_prjs -> per-batch effective weights/bias (tiny).
//   weff[b][c] = conv_w[c] + (silu(z) @ hwd_w + hwd_b) * (silu(z) @ c_w[0] + c_b[0])
//   beff[b]    = conv_b    + (silu(z) @ c_w[1] + c_b[1])
// ---------------------------------------------------------------------------
__global__ void zprj_weights_kernel(const float* __restrict__ z_prjs,
                                    const float* __restrict__ conv_w,
                                    const float* __restrict__ conv_b,
                                    const float* __restrict__ hwd_w,
                                    const float* __restrict__ hwd_b,
                                    const float* __restrict__ c_w,
                                    const float* __restrict__ c_b,
                                    float* __restrict__ weff,
                                    float* __restrict__ beff) {
    int b = threadIdx.x;
    if (b >= BATCH) return;
    float s[5];
    #pragma unroll
    for (int i = 0; i < 5; ++i) {
        float z = z_prjs[b * 5 + i];
        s[i] = z / (1.0f + __expf(-z));          // silu
    }
    float wb = hwd_b[0];
    float sc = c_b[0];
    float sh = c_b[1];
    #pragma unroll
    for (int i = 0; i < 5; ++i) {
        wb += s[i] * hwd_w[i];
        sc += s[i] * c_w[i];                     // c_w row 0
        sh += s[i] * c_w[5 + i];                 // c_w row 1
    }
    float ws = wb * sc;                          // scalar added to every channel weight
    #pragma unroll
    for (int c = 0; c < CH; ++c)
        weff[b * CH + c] = conv_w[c] + ws;
    beff[b] = conv_b[0] + sh;
}

// ---------------------------------------------------------------------------
// Kernel 2: streaming channel-mix via chained V_WMMA_F32_16X16X4_F32,
// two 16-wide spatial tiles per wave-iteration.
//
// D(16x16) = Sum_{4 K-chunks} A(16x4: weff, rows replicated) x
//            B(4x16: x[channel, spatial]); C seeded inline-0.
// A's rows are identical, so EVERY lane's accumulator VGPR0 holds
// out[lane%16] for its tile. Lanes 0-15 emit tile A, lanes 16-31 emit
// tile B (one v_cndmask), giving a single full-width coalesced NT store
// per iteration and no EXEC toggling in the loop (WMMA wants EXEC==~0).
// ---------------------------------------------------------------------------
__global__ void channel_mix_wmma_kernel(const float* __restrict__ x,
                                        const float* __restrict__ weff,
                                        const float* __restrict__ beff,
                                        float* __restrict__ out) {
    const int  lane   = threadIdx.x & 31;
    const int  half   = lane >> 4;     // 0: lanes 0-15, 1: lanes 16-31
    const int  n      = lane & 15;     // column / spatial-within-tile
    const int  wave   = (int)((blockIdx.x * blockDim.x + threadIdx.x) >> 5);
    const int  nwaves = (int)((gridDim.x * blockDim.x) >> 5);
    const unsigned step = (unsigned)nwaves * 32u;

    for (int b = 0; b < BATCH; ++b) {
        // A operand chunks (channels 4i..4i+3). 16x4 fp32 A layout:
        //   VGPR0 = K0 (lanes 0-15) / K2 (lanes 16-31); VGPR1 = K1 / K3.
        const float* wb = weff + b * CH;
        v2f a0, a1, a2, a3;
        a0.x = wb[ 0 + 2 * half]; a0.y = wb[ 1 + 2 * half];
        a1.x = wb[ 4 + 2 * half]; a1.y = wb[ 5 + 2 * half];
        a2.x = wb[ 8 + 2 * half]; a2.y = wb[ 9 + 2 * half];
        a3.x = wb[12 + 2 * half]; a3.y = wb[13 + 2 * half];
        const float bias = beff[b];

        // Wave-uniform chunk base pointers; all 16 loads per iteration use
        // these 4 bases + imm offsets {0, 64, DHW*4, DHW*4+64} (24-bit ok).
        const float* xb = x + (size_t)b * CH * DHW;
        const float* c0 = xb;
        const float* c1 = xb + (size_t)4 * DHW;
        const float* c2 = xb + (size_t)8 * DHW;
        const float* c3 = xb + (size_t)12 * DHW;
        float*       ob = out + (size_t)b * DHW;

        // Load offset (tile A; tile B = +16) and full-width store offset.
        unsigned voff = (unsigned)wave * 32u + (unsigned)n
                      + (unsigned)half * (2u * DHW);
        unsigned soff = (unsigned)wave * 32u + (unsigned)lane;

        for (int t = wave; t < PAIRS; t += nwaves) {
            // B operands: VGPR0 = row K0/K2, VGPR1 = row K1/K3, N = lane&15.
            v2f pA0, pA1, pA2, pA3, pB0, pB1, pB2, pB3;
            pA0.x = __builtin_nontemporal_load(c0 + voff);
            pA0.y = __builtin_nontemporal_load(c0 + voff + DHW);
            pB0.x = __builtin_nontemporal_load(c0 + voff + 16);
            pB0.y = __builtin_nontemporal_load(c0 + voff + DHW + 16);
            pA1.x = __builtin_nontemporal_load(c1 + voff);
            pA1.y = __builtin_nontemporal_load(c1 + voff + DHW);
            pB1.x = __builtin_nontemporal_load(c1 + voff + 16);
            pB1.y = __builtin_nontemporal_load(c1 + voff + DHW + 16);
            pA2.x = __builtin_nontemporal_load(c2 + voff);
            pA2.y = __builtin_nontemporal_load(c2 + voff + DHW);
            pB2.x = __builtin_nontemporal_load(c2 + voff + 16);
            pB2.y = __builtin_nontemporal_load(c2 + voff + DHW + 16);
            pA3.x = __builtin_nontemporal_load(c3 + voff);
            pA3.y = __builtin_nontemporal_load(c3 + voff + DHW);
            pB3.x = __builtin_nontemporal_load(c3 + voff + 16);
            pB3.y = __builtin_nontemporal_load(c3 + voff + DHW + 16);

            v8f cA = {};   // constant zero -> inline 0 SRC2 on first WMMA
            v8f cB = {};
            cA = __builtin_amdgcn_wmma_f32_16x16x4_f32(false, a0, false, pA0,
                                                       (short)0, cA, false, false);
            cB = __builtin_amdgcn_wmma_f32_16x16x4_f32(false, a0, false, pB0,
                                                       (short)0, cB, false, false);
            cA = __builtin_amdgcn_wmma_f32_16x16x4_f32(false, a1, false, pA1,
                                                       (short)0, cA, false, false);
            cB = __builtin_amdgcn_wmma_f32_16x16x4_f32(false, a1, false, pB1,
                                                       (short)0, cB, false, false);
            cA = __builtin_amdgcn_wmma_f32_16x16x4_f32(false, a2, false, pA2,
                                                       (short)0, cA, false, false);
            cB = __builtin_amdgcn_wmma_f32_16x16x4_f32(false, a2, false, pB2,
                                                       (short)0, cB, false, false);
            cA = __builtin_amdgcn_wmma_f32_16x16x4_f32(false, a3, false, pA3,
                                                       (short)0, cA, false, false);
            cB = __builtin_amdgcn_wmma_f32_16x16x4_f32(false, a3, false, pB3,
                                                       (short)0, cB, false, false);

            // Lanes 0-15 -> tile A col n; lanes 16-31 -> tile B col n.
            float val = (half ? cB[0] : cA[0]) + bias;
            __builtin_nontemporal_store(val, ob + soff);

            voff += step;
            soff += step;
        }
    }
}

// ---------------------------------------------------------------------------
extern "C" void kernel_launch(void* const* d_in, const int* in_sizes, int n_in,
                              void* d_out, int out_size, void* d_ws, size_t ws_size,
                              hipStream_t stream) {
    const float* x      = (const float*)d_in[0];
    const float* z_prjs = (const float*)d_in[1];
    const float* conv_w = (const float*)d_in[2];
    const float* conv_b = (const float*)d_in[3];
    const float* hwd_w  = (const float*)d_in[4];
    const float* hwd_b  = (const float*)d_in[5];
    const float* c_w    = (const float*)d_in[6];
    const float* c_b    = (const float*)d_in[7];

    float* weff = (float*)d_ws;            // 8*16 floats
    float* beff = weff + BATCH * CH;       // 8 floats
    float* outp = (float*)d_out;

    zprj_weights_kernel<<<1, 32, 0, stream>>>(z_prjs, conv_w, conv_b,
                                              hwd_w, hwd_b, c_w, c_b,
                                              weff, beff);

    // 1728 blocks * 8 waves = 13824 waves; 27648 tile-pairs/batch -> exactly
    // 2 pairs per wave per batch, no tail.
    channel_mix_wmma_kernel<<<1728, 256, 0, stream>>>(x, weff, beff, outp);
}